// SinkhornSolver_29265907155513
// MI455X (gfx1250) — compile-verified
//
#include <hip/hip_runtime.h>
#include <math.h>

// ---------------------------------------------------------------------------
// Sinkhorn OT solver for MI455X (gfx1250, wave32).
//   x, y: [2048, 64] f32.  Output: [cost(1), pi(2048*2048)] f32.
// C is materialized once with fp32 WMMA into the pi region of d_out (16 MB,
// stays resident in the 192 MB L2), then 100 row/col logsumexp passes stream
// it from L2; final pass overwrites C with pi in place.
// ---------------------------------------------------------------------------

#define NN 2048
#define DD 64
#define TPB 256
#define ROW_CHUNKS 32              // 2048 rows / 32 = 64 rows per chunk
#define ROWS_PER_CHUNK 64
#define EPSV 0.1f
#define INV_EPS 10.0f
#define THRV 0.1f
// log(1/2048 + 1e-8)
#define LOGAB (-7.6245985f)

typedef __attribute__((ext_vector_type(2))) float v2f;
typedef __attribute__((ext_vector_type(8))) float v8f;

// ---------------- init: u=v=0, x2/y2 row norms, done=0 ----------------------
__global__ void sk_init(const float* __restrict__ x, const float* __restrict__ y,
                        float* __restrict__ u, float* __restrict__ v,
                        float* __restrict__ x2, float* __restrict__ y2,
                        int* __restrict__ done) {
  int i = blockIdx.x * TPB + threadIdx.x;
  if (i < NN) {
    u[i] = 0.0f;
    v[i] = 0.0f;
    float sx = 0.0f, sy = 0.0f;
#pragma unroll
    for (int d = 0; d < DD; ++d) {
      float a = x[i * DD + d];
      float b = y[i * DD + d];
      sx = fmaf(a, a, sx);
      sy = fmaf(b, b, sy);
    }
    x2[i] = sx;
    y2[i] = sy;
  }
  if (i == 0) *done = 0;
}

// ---------------- C = x2_i + y2_j - 2 * (x @ y^T) via f32 WMMA --------------
// One wave per 16x16 output tile. K=64 -> 16 chained V_WMMA_F32_16X16X4_F32.
// A frag (16x4 f32): lanes 0-15 = rows M, vgpr{0,1} = K{0,1}; lanes 16-31 = K{2,3}.
// B frag (4x16 f32): lanes 0-15 = cols N, vgpr{0,1} = K{0,1}; lanes 16-31 = K{2,3}.
__global__ void sk_build_c(const float* __restrict__ x, const float* __restrict__ y,
                           const float* __restrict__ x2, const float* __restrict__ y2,
                           float* __restrict__ C) {
  int wid  = (blockIdx.x * TPB + threadIdx.x) >> 5;   // global wave id = tile id
  int lane = threadIdx.x & 31;
  int ti = wid >> 7;          // row tile (0..127)
  int tj = wid & 127;         // col tile (0..127)
  int lane_lo = lane & 15;
  int lhi = lane >> 4;        // 0: K pair {0,1}, 1: K pair {2,3}

  const float* xp = x + (ti * 16 + lane_lo) * DD + 2 * lhi;
  const float* yp = y + (tj * 16 + lane_lo) * DD + 2 * lhi;

  v8f acc = {};
#pragma unroll
  for (int kk = 0; kk < 16; ++kk) {
    v2f a = *(const v2f*)(xp + kk * 4);
    v2f b = *(const v2f*)(yp + kk * 4);
    acc = __builtin_amdgcn_wmma_f32_16x16x4_f32(
        /*neg_a=*/false, a, /*neg_b=*/false, b,
        /*c_mod=*/(short)0, acc, /*reuse_a=*/false, /*reuse_b=*/false);
  }

  // C/D layout: vgpr r, lanes 0-15 -> M=r, lanes 16-31 -> M=r+8; N = lane_lo.
  int j = tj * 16 + lane_lo;
  float y2j = y2[j];
#pragma unroll
  for (int r = 0; r < 8; ++r) {
    int i = ti * 16 + r + 8 * lhi;
    C[i * NN + j] = x2[i] + y2j - 2.0f * acc[r];
  }
}

// -------- online LSE pair combine: (m,s) <- (m,s) (+) (m2,s2) ---------------
__device__ inline void lse_acc(float& m, float& s, float t) {
  float M = fmaxf(m, t);
  s = s * __expf(m - M) + __expf(t - M);
  m = M;
}
__device__ inline void lse_merge(float& m, float& s, float m2, float s2) {
  float M = fmaxf(m, m2);
  s = s * __expf(m - M) + s2 * __expf(m2 - M);
  m = M;
}

// ---------------- u_new_i = eps*log_a - eps*LSE_j((v_j - C_ij)/eps) ---------
__global__ void sk_row_lse(const float* __restrict__ C, const float* __restrict__ v,
                           float* __restrict__ un, const int* __restrict__ done) {
  if (*done) return;
  int i = blockIdx.x;
  int t = threadIdx.x;
  const float* Crow = C + (size_t)i * NN;

  float m = -INFINITY, s = 0.0f;
#pragma unroll
  for (int k = 0; k < NN / TPB; ++k) {
    int j = t + k * TPB;
    lse_acc(m, s, (v[j] - Crow[j]) * INV_EPS);
  }

  __shared__ float smx[TPB], sms[TPB];
  smx[t] = m; sms[t] = s;
  __syncthreads();
  for (int w = TPB / 2; w > 0; w >>= 1) {
    if (t < w) {
      float mm = smx[t], ss = sms[t];
      lse_merge(mm, ss, smx[t + w], sms[t + w]);
      smx[t] = mm; sms[t] = ss;
    }
    __syncthreads();
  }
  if (t == 0) {
    float lse = smx[0] + __logf(sms[0]);
    un[i] = EPSV * LOGAB - EPSV * lse;
  }
}

// ------- column partial LSE over a 64-row chunk (coalesced: lane = column) --
__global__ void sk_col_partial(const float* __restrict__ C, const float* __restrict__ un,
                               float* __restrict__ pm, float* __restrict__ ps,
                               const int* __restrict__ done) {
  if (*done) return;
  int j  = blockIdx.x * TPB + threadIdx.x;   // column
  int rc = blockIdx.y;                       // row chunk
  int i0 = rc * ROWS_PER_CHUNK;

  float m = -INFINITY, s = 0.0f;
#pragma unroll 4
  for (int r = 0; r < ROWS_PER_CHUNK; ++r) {
    int i = i0 + r;
    lse_acc(m, s, (un[i] - C[(size_t)i * NN + j]) * INV_EPS);
  }
  pm[rc * NN + j] = m;
  ps[rc * NN + j] = s;
}

// ---------------- v_new_j from the 32 partials (fixed order) ----------------
__global__ void sk_col_combine(const float* __restrict__ pm, const float* __restrict__ ps,
                               float* __restrict__ vn, const int* __restrict__ done) {
  if (*done) return;
  int j = blockIdx.x * TPB + threadIdx.x;
  float m = pm[j], s = ps[j];
#pragma unroll
  for (int rc = 1; rc < ROW_CHUNKS; ++rc)
    lse_merge(m, s, pm[rc * NN + j], ps[rc * NN + j]);
  vn[j] = EPSV * LOGAB - EPSV * (m + __logf(s));
}

// --------- commit: diff = sum|un-u|+|vn-v| (deterministic), copy, done ------
__global__ void sk_commit(float* __restrict__ u, float* __restrict__ v,
                          const float* __restrict__ un, const float* __restrict__ vn,
                          int* __restrict__ done) {
  if (*done) return;
  int t = threadIdx.x;
  float local = 0.0f;
#pragma unroll
  for (int k = 0; k < NN / TPB; ++k) {
    int i = t + k * TPB;
    local += fabsf(un[i] - u[i]) + fabsf(vn[i] - v[i]);
  }
  __shared__ float sm[TPB];
  sm[t] = local;
  __syncthreads();
  for (int w = TPB / 2; w > 0; w >>= 1) {
    if (t < w) sm[t] += sm[t + w];
    __syncthreads();
  }
  float diff = sm[0];
  // commit this iteration's values (reference commits before updating `done`)
#pragma unroll
  for (int k = 0; k < NN / TPB; ++k) {
    int i = t + k * TPB;
    u[i] = un[i];
    v[i] = vn[i];
  }
  if (t == 0 && diff < THRV) *done = 1;
}

// ---------------- pi = exp((u+v-C)/eps) in place; partial cost sums ---------
__global__ void sk_final_pi(float* __restrict__ C, const float* __restrict__ u,
                            const float* __restrict__ v, float* __restrict__ partial) {
  size_t idx = (size_t)blockIdx.x * TPB + threadIdx.x;
  int i = (int)(idx >> 11);
  int j = (int)(idx & (NN - 1));
  float c = C[idx];
  float p = __expf((u[i] + v[j] - c) * INV_EPS);
  C[idx] = p;

  __shared__ float sm[TPB];
  sm[threadIdx.x] = p * c;
  __syncthreads();
  for (int w = TPB / 2; w > 0; w >>= 1) {
    if (threadIdx.x < w) sm[threadIdx.x] += sm[threadIdx.x + w];
    __syncthreads();
  }
  if (threadIdx.x == 0) partial[blockIdx.x] = sm[0];
}

__global__ void sk_final_reduce(const float* __restrict__ partial, float* __restrict__ out) {
  int t = threadIdx.x;
  float local = 0.0f;
  for (int k = t; k < (NN * NN) / TPB; k += TPB) local += partial[k];
  __shared__ float sm[TPB];
  sm[t] = local;
  __syncthreads();
  for (int w = TPB / 2; w > 0; w >>= 1) {
    if (t < w) sm[t] += sm[t + w];
    __syncthreads();
  }
  if (t == 0) out[0] = sm[0];
}

// ---------------------------------------------------------------------------
extern "C" void kernel_launch(void* const* d_in, const int* in_sizes, int n_in,
                              void* d_out, int out_size, void* d_ws, size_t ws_size,
                              hipStream_t stream) {
  (void)in_sizes; (void)n_in; (void)out_size; (void)ws_size;
  const float* x = (const float*)d_in[0];
  const float* y = (const float*)d_in[1];
  float* out = (float*)d_out;
  float* C   = out + 1;                 // pi region doubles as C storage

  float* W  = (float*)d_ws;
  float* u  = W;                        // 2048
  float* v  = W + 2048;                 // 2048
  float* un = W + 4096;                 // 2048
  float* vn = W + 6144;                 // 2048
  float* x2 = W + 8192;                 // 2048
  float* y2 = W + 10240;                // 2048
  float* pm = W + 12288;                // 32*2048
  float* ps = W + 12288 + ROW_CHUNKS * NN;            // 32*2048
  float* partial = ps + ROW_CHUNKS * NN;              // 16384
  int*   done = (int*)(partial + (NN * NN) / TPB);

  sk_init<<<NN / TPB, TPB, 0, stream>>>(x, y, u, v, x2, y2, done);
  sk_build_c<<<(128 * 128) / 8, TPB, 0, stream>>>(x, y, x2, y2, C);

  for (int it = 0; it < 100; ++it) {
    sk_row_lse<<<NN, TPB, 0, stream>>>(C, v, un, done);
    sk_col_partial<<<dim3(NN / TPB, ROW_CHUNKS), TPB, 0, stream>>>(C, un, pm, ps, done);
    sk_col_combine<<<NN / TPB, TPB, 0, stream>>>(pm, ps, vn, done);
    sk_commit<<<1, TPB, 0, stream>>>(u, v, un, vn, done);
  }

  sk_final_pi<<<(NN * NN) / TPB, TPB, 0, stream>>>(C, u, v, partial);
  sk_final_reduce<<<1, TPB, 0, stream>>>(partial, out);
}